// CrossOp_29583734735061
// MI455X (gfx1250) — compile-verified
//
#include <hip/hip_runtime.h>

typedef float v2f __attribute__((ext_vector_type(2)));
typedef float v8f __attribute__((ext_vector_type(8)));

#if defined(__has_builtin)
#if __has_builtin(__builtin_amdgcn_global_load_async_to_lds_b32) && \
    __has_builtin(__builtin_amdgcn_s_wait_asynccnt)
#define USE_ASYNC_LDS 1
#endif
#endif
#ifndef USE_ASYNC_LDS
#define USE_ASYNC_LDS 0
#endif

#define C_IN 64
#define HW   64
#define PADW 66                       // 64 cols + halo
#define ROWS_PER_WG 4                 // output rows per workgroup
#define IN_ROWS 6                     // input rows incl. halo
#define IN_ELEMS (IN_ROWS * C_IN * PADW)   // 25344 floats (99 KB)
#define W_ELEMS  (9 * 64 * 64)             // 36864 floats (144 KB)
#define SMEM_BYTES ((IN_ELEMS + W_ELEMS) * 4)  // ~243 KB of the WGP's 320 KB LDS

__device__ __forceinline__ void stage_b32(const float* g, float* l)
{
#if USE_ASYNC_LDS
    __builtin_amdgcn_global_load_async_to_lds_b32(
        (__attribute__((address_space(1))) int*)(g),
        (__attribute__((address_space(3))) int*)(l), 0, 0);
#else
    *l = *g;
#endif
}

// One WG (512 thr = 16 waves) computes a 64(co) x 4(rows) x 64(px) output tile.
// Each wave owns a 2Mx2N register block (4 accumulators); every WMMA operand is
// a single aligned ds_load_b64 with an immediate offset from a per-tap base.
// Input LDS layout is channel-pair interleaved: [row][c/2][x][c&1] so the two
// B components per lane are contiguous 8 bytes.
// mode 0: conv of target (weight channels [0,64)), write raw conv.
// mode 1: conv of support (channels [64,128)), write conv + ct + bias.
__global__ __launch_bounds__(512) void conv_cross_wmma_kernel(
    const float* __restrict__ input,
    const float* __restrict__ weight,   // (64, 128, 3, 3)
    const float* __restrict__ bias,
    const float* __restrict__ ct,
    float* __restrict__ out,
    int mode)
{
    extern __shared__ float smem[];
    float* in_lds = smem;               // [6][32 cpair][66 x][2]
    float* w_lds  = smem + IN_ELEMS;    // [9 tap][64 co][64 c]

    const int t    = threadIdx.x;
    const int lane = t & 31;
    const int wv   = t >> 5;            // 0..15
    const int l16  = lane & 15;
    const int hi   = (lane >> 4) & 1;

    const int mpair = wv & 1;           // co halves: 0-31 / 32-63
    const int npair = wv >> 1;          // 0..7 -> 32-px strip
    const int ro    = npair >> 1;       // output row in group (0..3)
    const int xb    = (npair & 1) * 32; // pixel column base

    const int hg  = blockIdx.x * ROWS_PER_WG;
    const int img = blockIdx.y;
    const float* imgp = input + img * (C_IN * HW * HW);
    const int cOff = (mode == 0) ? 0 : 64;

    // ---- stage input rows hg-1 .. hg+4 (zero halo), interleaved layout ----
    for (int idx = t; idx < IN_ELEMS; idx += 512) {
        int r   = idx / (C_IN * PADW);
        int rem = idx - r * (C_IN * PADW);
        int c   = rem / PADW;
        int x1  = rem - c * PADW;
        int gy  = hg + r - 1;
        int gx  = x1 - 1;
        float* dst = &in_lds[((r * 32 + (c >> 1)) * PADW + x1) * 2 + (c & 1)];
        if ((unsigned)gy < HW && (unsigned)gx < HW)
            stage_b32(&imgp[(c * HW + gy) * HW + gx], dst);
        else
            *dst = 0.0f;
    }
    // ---- stage ALL weights of this branch: w_lds[tap][co][c] ----
    for (int j = t; j < W_ELEMS; j += 512) {
        int co  = j / 576;
        int r   = j - co * 576;         // (c*9 + tap)
        int c   = r / 9;
        int tap = r - c * 9;
        stage_b32(&weight[co * 1152 + cOff * 9 + r],
                  &w_lds[(tap * 64 + co) * 64 + c]);
    }
#if USE_ASYNC_LDS
    __builtin_amdgcn_s_wait_asynccnt(0);
#endif
    __syncthreads();                    // the only barrier

    v8f acc00 = {}, acc01 = {}, acc10 = {}, acc11 = {};

    for (int tap = 0; tap < 9; ++tap) {
        const int ky = tap / 3;
        const int dx = tap - ky * 3 - 1;            // kx-1
        const int col0 = xb + l16 + dx + 1;
        // A base: [tap][mpair*32 + l16][hi*2]; second M tile at +1024 floats
        const float* abase = &w_lds[(tap * 64 + mpair * 32 + l16) * 64 + hi * 2];
        // B base: row ro+ky, channel pair 'hi', column col0 (interleaved pairs)
        const float* bbase = &in_lds[(((ro + ky) * 32 + hi) * PADW + col0) * 2];

#pragma unroll
        for (int s = 0; s < 16; ++s) {
            v2f a0 = *(const v2f*)(abase + s * 4);           // ch k0..k0+1 (+hi*2)
            v2f a1 = *(const v2f*)(abase + 1024 + s * 4);    // M tile 1
            v2f b0 = *(const v2f*)(bbase + s * (2 * PADW * 2));        // col0
            v2f b1 = *(const v2f*)(bbase + s * (2 * PADW * 2) + 32);   // col0+16
            acc00 = __builtin_amdgcn_wmma_f32_16x16x4_f32(false, a0, false, b0, (short)0, acc00, false, false);
            acc01 = __builtin_amdgcn_wmma_f32_16x16x4_f32(false, a0, false, b1, (short)0, acc01, false, false);
            acc10 = __builtin_amdgcn_wmma_f32_16x16x4_f32(false, a1, false, b0, (short)0, acc10, false, false);
            acc11 = __builtin_amdgcn_wmma_f32_16x16x4_f32(false, a1, false, b1, (short)0, acc11, false, false);
        }
    }

    // ---- epilogue: D VGPR rr -> (M = rr + hi*8, N = lane%16) ----
    const int h  = hg + ro;
    const int x0 = xb + l16;
    const int obase = img * (64 * HW * HW) + h * HW;
    if (mode == 0) {
#pragma unroll
        for (int rr = 0; rr < 8; ++rr) {
            int co0 = mpair * 32 + rr + hi * 8;
            int co1 = co0 + 16;
            out[obase + co0 * 4096 + x0]      = acc00[rr];
            out[obase + co0 * 4096 + x0 + 16] = acc01[rr];
            out[obase + co1 * 4096 + x0]      = acc10[rr];
            out[obase + co1 * 4096 + x0 + 16] = acc11[rr];
        }
    } else {
        const int b  = img >> 4;
        const int cb = b * (64 * HW * HW) + h * HW;
#pragma unroll
        for (int rr = 0; rr < 8; ++rr) {
            int co0 = mpair * 32 + rr + hi * 8;
            int co1 = co0 + 16;
            float bi0 = bias[co0], bi1 = bias[co1];
            out[obase + co0 * 4096 + x0]      = acc00[rr] + ct[cb + co0 * 4096 + x0]      + bi0;
            out[obase + co0 * 4096 + x0 + 16] = acc01[rr] + ct[cb + co0 * 4096 + x0 + 16] + bi0;
            out[obase + co1 * 4096 + x0]      = acc10[rr] + ct[cb + co1 * 4096 + x0]      + bi1;
            out[obase + co1 * 4096 + x0 + 16] = acc11[rr] + ct[cb + co1 * 4096 + x0 + 16] + bi1;
        }
    }
}

// aggregated[b,0,co,h,w] = mean_s interactions[b,s,co,h,w]
__global__ __launch_bounds__(256) void mean_over_support_kernel(
    const float* __restrict__ inter, float* __restrict__ aggr)
{
    int i = blockIdx.x * blockDim.x + threadIdx.x;   // 0..524287
    if (i < 2 * 64 * HW * HW) {
        int b = i >> 18;
        int r = i & 262143;
        const float* p = inter + b * 16 * 262144 + r;
        float s = 0.0f;
#pragma unroll
        for (int j = 0; j < 16; ++j) s += p[j * 262144];
        aggr[i] = s * 0.0625f;
    }
}

extern "C" void kernel_launch(void* const* d_in, const int* in_sizes, int n_in,
                              void* d_out, int out_size, void* d_ws, size_t ws_size,
                              hipStream_t stream)
{
    (void)in_sizes; (void)n_in; (void)out_size; (void)d_ws; (void)ws_size;
    const float* target  = (const float*)d_in[0];  // (2,1,64,64,64)
    const float* support = (const float*)d_in[1];  // (2,16,64,64,64)
    const float* weight  = (const float*)d_in[2];  // (64,128,3,3)
    const float* bias    = (const float*)d_in[3];  // (64)

    float* out   = (float*)d_out;
    float* ctbuf = out;                    // aggregated region doubles as scratch
    float* inter = out + 2 * 64 * 64 * 64;

    conv_cross_wmma_kernel<<<dim3(HW / ROWS_PER_WG, 2), 512, SMEM_BYTES, stream>>>(
        target, weight, bias, nullptr, ctbuf, 0);
    conv_cross_wmma_kernel<<<dim3(HW / ROWS_PER_WG, 32), 512, SMEM_BYTES, stream>>>(
        support, weight, bias, ctbuf, inter, 1);
    mean_over_support_kernel<<<2048, 256, 0, stream>>>(inter, out);
}